// FuzzyContrastiveLoss_45329084842231
// MI455X (gfx1250) — compile-verified
//
#include <hip/hip_runtime.h>
#include <math.h>

typedef __attribute__((ext_vector_type(2))) float v2f;
typedef __attribute__((ext_vector_type(8))) float v8f;

#define L_TOTAL 8192
#define C_CLUST 64
#define DDIM    64
#define ROWS_PER_BLOCK 128
#define PAD     68   // LDS row stride in words: bank = (4*row + k) % 64 -> conflict-free b64 loads

// ---------------------------------------------------------------------------
// Kernel 1: batched GEMM  y[l,:] = (z[l]-mu[c]) @ A[c]   via V_WMMA_F32_16X16X4_F32
//           d2[l,c] = ||y||^2 (rowwise),  dd[l,c] = ||z[l]-mu[c]||^2
// Outputs stored transposed (c-major) for coalesced column processing later.
// ---------------------------------------------------------------------------
__global__ __launch_bounds__(256)
void k1_d2_dd(const float* __restrict__ z, const float* __restrict__ mu,
              const float* __restrict__ sig,
              float* __restrict__ d2T, float* __restrict__ ddT)
{
    // AsT[j*PAD + k] = A[c][k][j]  (transposed: K contiguous -> single b64 B loads)
    __shared__ __attribute__((aligned(16))) float AsT[DDIM * PAD];            // ~17 KB
    // dsh[r*PAD + k] = z[l0+r][k] - mu[c][k]  (pre-subtracted A-operand)
    __shared__ __attribute__((aligned(16))) float dsh[ROWS_PER_BLOCK * PAD];  // ~34 KB

    const int tid = threadIdx.x;
    const int c  = blockIdx.y;
    const int l0 = blockIdx.x * ROWS_PER_BLOCK;

    const float* Ac = sig + (size_t)c * DDIM * DDIM;
    for (int i = tid; i < DDIM * DDIM; i += 256) {
        const int k = i >> 6, j = i & 63;
        AsT[j * PAD + k] = Ac[i];                       // coalesced read, transpose in LDS
    }
    for (int i = tid; i < ROWS_PER_BLOCK * DDIM; i += 256) {
        const int r = i >> 6, k = i & 63;
        dsh[r * PAD + k] = z[(size_t)l0 * DDIM + i] - mu[c * DDIM + k];
    }
    __syncthreads();

    const int wave = tid >> 5;
    const int lane = tid & 31;
    const int m  = lane & 15;    // A row within tile / B-C-D column within j-tile
    const int hi = lane >> 4;    // half-wave select
    const int r0 = wave * 16;    // this wave's 16-row tile inside the block

    v8f acc0 = {}; v8f acc1 = {}; v8f acc2 = {}; v8f acc3 = {};
    float ddp = 0.0f;

    // K loop: 16 steps of K=4.  32-bit A 16x4 layout:
    // lanes 0-15 hold K = k0,k0+1 ; lanes 16-31 hold K = k0+2,k0+3.
#pragma unroll
    for (int k0 = 0; k0 < DDIM; k0 += 4) {
        const int ka = k0 + 2 * hi;
        const v2f a = *(const v2f*)&dsh[(r0 + m) * PAD + ka];
        ddp += a.x * a.x + a.y * a.y;

        const v2f b0 = *(const v2f*)&AsT[(m     ) * PAD + ka];
        const v2f b1 = *(const v2f*)&AsT[(m + 16) * PAD + ka];
        const v2f b2 = *(const v2f*)&AsT[(m + 32) * PAD + ka];
        const v2f b3 = *(const v2f*)&AsT[(m + 48) * PAD + ka];

        acc0 = __builtin_amdgcn_wmma_f32_16x16x4_f32(false, a, false, b0, (short)0, acc0, false, false);
        acc1 = __builtin_amdgcn_wmma_f32_16x16x4_f32(false, a, false, b1, (short)0, acc1, false, false);
        acc2 = __builtin_amdgcn_wmma_f32_16x16x4_f32(false, a, false, b2, (short)0, acc2, false, false);
        acc3 = __builtin_amdgcn_wmma_f32_16x16x4_f32(false, a, false, b3, (short)0, acc3, false, false);
    }

    // d2 per row: square-accumulate the 16x64 Y tile.  C/D layout: VGPR r holds
    // row M = r + 8*hi, column N = lane&15 (+16*t per accumulator).
    float s[8];
#pragma unroll
    for (int r = 0; r < 8; ++r)
        s[r] = acc0[r] * acc0[r] + acc1[r] * acc1[r] +
               acc2[r] * acc2[r] + acc3[r] * acc3[r];
#pragma unroll
    for (int mask = 1; mask <= 8; mask <<= 1) {
#pragma unroll
        for (int r = 0; r < 8; ++r)
            s[r] += __shfl_xor(s[r], mask, 32);
    }
    if (m == 0) {
        float* dst = d2T + (size_t)c * L_TOTAL + l0 + r0 + 8 * hi;
#pragma unroll
        for (int r = 0; r < 8; ++r) dst[r] = s[r];
    }

    // dd per row: half 0 covers K = {0,1} mod 4, half 1 covers {2,3}.
    const float ddfull = ddp + __shfl_xor(ddp, 16, 32);
    if (hi == 0)
        ddT[(size_t)c * L_TOTAL + l0 + r0 + m] = ddfull;
}

// ---------------------------------------------------------------------------
// Kernel 2: per-column softmax stats over the L axis.
//   colmin[c] = min_l d2[l,c];  colZinv[c] = 1 / sum_l exp(colmin - d2)
// ---------------------------------------------------------------------------
__global__ __launch_bounds__(256)
void k2_colstats(const float* __restrict__ d2T,
                 float* __restrict__ colmin, float* __restrict__ colZinv)
{
    __shared__ float red[256];
    const int c   = blockIdx.x;
    const int tid = threadIdx.x;
    const float* col = d2T + (size_t)c * L_TOTAL;

    float mn = 3.402823466e38f;
    for (int l = tid; l < L_TOTAL; l += 256) mn = fminf(mn, col[l]);
    red[tid] = mn; __syncthreads();
    for (int s = 128; s > 0; s >>= 1) {
        if (tid < s) red[tid] = fminf(red[tid], red[tid + s]);
        __syncthreads();
    }
    const float cmin = red[0];
    __syncthreads();

    float sum = 0.0f;
    for (int l = tid; l < L_TOTAL; l += 256) sum += expf(cmin - col[l]);
    red[tid] = sum; __syncthreads();
    for (int s = 128; s > 0; s >>= 1) {
        if (tid < s) red[tid] += red[tid + s];
        __syncthreads();
    }
    if (tid == 0) { colmin[c] = cmin; colZinv[c] = 1.0f / red[0]; }
}

// ---------------------------------------------------------------------------
// Kernel 3: per-row fused terms.
//   pos_l = sum_c psi*dd ;  neg_l = (C - sum_c psi) * sum_c relu(1-d2)
// ---------------------------------------------------------------------------
__global__ __launch_bounds__(256)
void k3_rows(const float* __restrict__ d2T, const float* __restrict__ ddT,
             const float* __restrict__ colmin, const float* __restrict__ colZinv,
             float* __restrict__ partials)
{
    const int l = blockIdx.x * 256 + threadIdx.x;
    float pos = 0.0f, rowpsi = 0.0f, relu = 0.0f;
    for (int c = 0; c < C_CLUST; ++c) {
        const float d2v = d2T[(size_t)c * L_TOTAL + l];
        const float psi = expf(colmin[c] - d2v) * colZinv[c];
        pos    += psi * ddT[(size_t)c * L_TOTAL + l];
        rowpsi += psi;
        relu   += fmaxf(1.0f - d2v, 0.0f);
    }
    float v = pos + ((float)C_CLUST - rowpsi) * relu;

    __shared__ float red[256];
    red[threadIdx.x] = v; __syncthreads();
    for (int s = 128; s > 0; s >>= 1) {
        if (threadIdx.x < s) red[threadIdx.x] += red[threadIdx.x + s];
        __syncthreads();
    }
    if (threadIdx.x == 0) partials[blockIdx.x] = red[0];
}

// ---------------------------------------------------------------------------
// Kernel 4: final reduction of 32 block partials, divide by L.
// ---------------------------------------------------------------------------
__global__ __launch_bounds__(32)
void k4_final(const float* __restrict__ partials, float* __restrict__ out)
{
    float v = partials[threadIdx.x];   // exactly 32 partials, one wave
#pragma unroll
    for (int mask = 16; mask >= 1; mask >>= 1) v += __shfl_xor(v, mask, 32);
    if (threadIdx.x == 0) out[0] = v / (float)L_TOTAL;
}

// ---------------------------------------------------------------------------
extern "C" void kernel_launch(void* const* d_in, const int* in_sizes, int n_in,
                              void* d_out, int out_size, void* d_ws, size_t ws_size,
                              hipStream_t stream)
{
    (void)in_sizes; (void)n_in; (void)out_size; (void)ws_size;
    const float* z   = (const float*)d_in[0];   // (8192,1,64)
    const float* mu  = (const float*)d_in[1];   // (64,64)
    const float* sig = (const float*)d_in[2];   // (64,64,64)
    float* out = (float*)d_out;

    float* w        = (float*)d_ws;
    float* d2T      = w;                                   // 64*8192 floats
    float* ddT      = w + (size_t)C_CLUST * L_TOTAL;       // 64*8192 floats
    float* colmin   = w + 2 * (size_t)C_CLUST * L_TOTAL;   // 64
    float* colZinv  = colmin + C_CLUST;                    // 64
    float* partials = colZinv + C_CLUST;                   // 32

    dim3 g1(L_TOTAL / ROWS_PER_BLOCK, C_CLUST);            // (64, 64)
    k1_d2_dd<<<g1, 256, 0, stream>>>(z, mu, sig, d2T, ddT);
    k2_colstats<<<C_CLUST, 256, 0, stream>>>(d2T, colmin, colZinv);
    k3_rows<<<L_TOTAL / 256, 256, 0, stream>>>(d2T, ddT, colmin, colZinv, partials);
    k4_final<<<1, 32, 0, stream>>>(partials, out);
}